// RnnBlock_42571715838471
// MI455X (gfx1250) — compile-verified
//
#include <hip/hip_runtime.h>
#include <math.h>

// GRU stack: B=64, I=320, H=1536, L=3, O=256
// fp32 matrix math via V_WMMA_F32_16X16X4_F32 (exact fp32, matches reference).
// Each wave computes a 16x64 output strip (4 N-tiles) so one A-load feeds 4 WMMAs.

typedef __attribute__((ext_vector_type(2))) float v2f;
typedef __attribute__((ext_vector_type(8))) float v8f;

#define Bsz  64
#define Hsz  1536
#define H3   4608
#define BH   (Bsz * Hsz)   // 98304
#define BH3  (Bsz * H3)    // 294912
#define NT   4             // N-tiles per wave (16x64 strip)

// One wave32 computes a 16x(16*NT) strip of D = A[M,K] @ W[N,K]^T + bias[N].
// gridDim.z==2 selects a second, independent problem (gi/gh fused dual-GEMM).
__global__ __launch_bounds__(32) void gemm_bias_wmma_f32(
    const float* __restrict__ A0, const float* __restrict__ W0,
    const float* __restrict__ bias0, float* __restrict__ D0,
    const float* __restrict__ A1, const float* __restrict__ W1,
    const float* __restrict__ bias1, float* __restrict__ D1,
    int M, int N, int K)
{
    const float* __restrict__ A    = (blockIdx.z == 0) ? A0    : A1;
    const float* __restrict__ W    = (blockIdx.z == 0) ? W0    : W1;
    const float* __restrict__ bias = (blockIdx.z == 0) ? bias0 : bias1;
    float*       __restrict__ D    = (blockIdx.z == 0) ? D0    : D1;

    const int lane  = threadIdx.x;        // 0..31, wave32
    const int idx16 = lane & 15;          // row (A) / col (B) within tile
    const int half  = lane >> 4;          // selects K pair {2*half, 2*half+1}
    const int n0 = blockIdx.x * (16 * NT);
    const int m0 = blockIdx.y * 16;

    // A 16x4 f32 layout: lane<16 holds (M=lane, K=0..1), lane>=16 (M=lane-16, K=2..3)
    // B  4x16 f32 layout: lane<16 holds (N=lane, K=0..1), lane>=16 (N=lane-16, K=2..3)
    const v2f* __restrict__ ap = reinterpret_cast<const v2f*>(A + (size_t)(m0 + idx16) * K);
    const v2f* wp[NT];
#pragma unroll
    for (int t = 0; t < NT; ++t)
        wp[t] = reinterpret_cast<const v2f*>(W + (size_t)(n0 + t * 16 + idx16) * K);

    v8f acc[NT];
#pragma unroll
    for (int t = 0; t < NT; ++t) acc[t] = (v8f){};

#pragma unroll 4
    for (int k = 0; k < K; k += 4) {
        const int ki = (k >> 1) + half;
        v2f a = ap[ki];                  // one A-load feeds NT WMMAs
#pragma unroll
        for (int t = 0; t < NT; ++t) {
            v2f b = wp[t][ki];
            // (neg_a, A, neg_b, B, c_mod, C, reuse_a, reuse_b)
            acc[t] = __builtin_amdgcn_wmma_f32_16x16x4_f32(
                         false, a, false, b, (short)0, acc[t], false, false);
        }
    }

    // C/D 16x16 f32 layout: VGPR r -> row (half*8 + r), lane&15 -> column
#pragma unroll
    for (int t = 0; t < NT; ++t) {
        const int nc = n0 + t * 16 + idx16;
        const float bn = bias[nc];
        float* __restrict__ drow = D + (size_t)(m0 + half * 8) * N + nc;
#pragma unroll
        for (int r = 0; r < 8; ++r)
            drow[(size_t)r * N] = acc[t][r] + bn;
    }
}

// GRU gate fusion: h = (1-z)*n + z*h_prev,  gate order r,z,n
__global__ __launch_bounds__(256) void gru_gate(
    const float* __restrict__ gi, const float* __restrict__ gh,
    const float* __restrict__ hprev, float* __restrict__ hout)
{
    int idx = blockIdx.x * 256 + threadIdx.x;
    if (idx >= BH) return;
    int b = idx / Hsz;
    int j = idx - b * Hsz;
    const float* gib = gi + (size_t)b * H3;
    const float* ghb = gh + (size_t)b * H3;

    float ir = gib[j], iz = gib[j + Hsz], in_ = gib[j + 2 * Hsz];
    float hr = ghb[j], hz = ghb[j + Hsz], hn  = ghb[j + 2 * Hsz];

    float r = 1.0f / (1.0f + __expf(-(ir + hr)));
    float z = 1.0f / (1.0f + __expf(-(iz + hz)));
    float n = tanhf(in_ + r * hn);
    float hp = hprev[idx];
    hout[idx] = (1.0f - z) * n + z * hp;
}

extern "C" void kernel_launch(void* const* d_in, const int* in_sizes, int n_in,
                              void* d_out, int out_size, void* d_ws, size_t ws_size,
                              hipStream_t stream)
{
    const float* x      = (const float*)d_in[0];  // [64,320]
    const float* hidden = (const float*)d_in[1];  // [3,64,1536]
    const float* w_i2h  = (const float*)d_in[2];  // [1536,320]
    const float* b_i2h  = (const float*)d_in[3];  // [1536]
    const float* w_ih   = (const float*)d_in[4];  // [3,4608,1536]
    const float* w_hh   = (const float*)d_in[5];  // [3,4608,1536]
    const float* b_ih   = (const float*)d_in[6];  // [3,4608]
    const float* b_hh   = (const float*)d_in[7];  // [3,4608]
    const float* w_h2o  = (const float*)d_in[8];  // [256,1536]
    const float* b_h2o  = (const float*)d_in[9];  // [256]

    float* out     = (float*)d_out;               // [64,256]
    float* hid_out = out + Bsz * 256;             // [3,64,1536]

    float* ws  = (float*)d_ws;
    float* inp = ws;                              // 98304 floats
    float* gi  = ws + BH;                         // 294912 floats
    float* gh  = gi + BH3;                        // 294912 floats

    const dim3 wave(32);

    // inp = x @ w_i2h^T + b_i2h      [64,1536], K=320
    gemm_bias_wmma_f32<<<dim3(Hsz / (16 * NT), Bsz / 16, 1), wave, 0, stream>>>(
        x, w_i2h, b_i2h, inp,  x, w_i2h, b_i2h, inp,  Bsz, Hsz, 320);

    const float* cur = inp;
    for (int l = 0; l < 3; ++l) {
        const float* hp = hidden + (size_t)l * BH;
        // Fused dual GEMM: z=0 -> gi = cur @ w_ih^T + b_ih
        //                  z=1 -> gh = hp  @ w_hh^T + b_hh
        gemm_bias_wmma_f32<<<dim3(H3 / (16 * NT), Bsz / 16, 2), wave, 0, stream>>>(
            cur, w_ih + (size_t)l * H3 * Hsz, b_ih + (size_t)l * H3, gi,
            hp,  w_hh + (size_t)l * H3 * Hsz, b_hh + (size_t)l * H3, gh,
            Bsz, H3, Hsz);

        float* hq = hid_out + (size_t)l * BH;
        gru_gate<<<dim3((BH + 255) / 256), dim3(256), 0, stream>>>(gi, gh, hp, hq);
        cur = hq;
    }

    // output = h2 @ w_h2o^T + b_h2o  [64,256], K=1536
    gemm_bias_wmma_f32<<<dim3(256 / (16 * NT), Bsz / 16, 1), wave, 0, stream>>>(
        cur, w_h2o, b_h2o, out,  cur, w_h2o, b_h2o, out,  Bsz, 256, Hsz);
}